// GCNRelationPredictor_67894843015673
// MI455X (gfx1250) — compile-verified
//
#include <hip/hip_runtime.h>

// ---------------------------------------------------------------------------
// 2-layer GCN (PyG GCNConv) for MI455X / gfx1250, fp32 end-to-end.
//   deg/dinv -> edge prep (int32 idx + fused norm) -> hw1 = x@W1 (WMMA)
//   -> atomic scatter (L2-resident) -> +selfloop+bias+ReLU
//   -> hw2 = h@W2 (WMMA) -> scatter -> finalize.
// GEMMs: V_WMMA_F32_16X16X4_F32 (wave32); B panel pre-swizzled in LDS so a
// B fragment is one conflict-free ds_load_b64 (compiler pairs them into
// ds_load_2addr_b64). A-stream prefetched with global_prefetch_b8.
// ---------------------------------------------------------------------------

typedef __attribute__((ext_vector_type(2))) float v2f;
typedef __attribute__((ext_vector_type(8))) float v8f;

#define N_NODES 100000
#define N_EDGES 1600000
#define IN_DIM 256
#define HID_DIM 128
#define OUT_DIM 237
#define M_STRIPS (N_NODES / 16)  // 6250

__global__ void init_ones_kernel(float* __restrict__ p, int n) {
  int i = blockIdx.x * blockDim.x + threadIdx.x;
  if (i < n) p[i] = 1.0f;  // self-loop contributes 1 to every degree
}

__global__ void degree_kernel(const long long* __restrict__ ei,
                              float* __restrict__ deg, int E) {
  int e = blockIdx.x * blockDim.x + threadIdx.x;
  if (e < E) atomicAdd(&deg[ei[(size_t)E + e]], 1.0f);
}

__global__ void rsqrt_kernel(float* __restrict__ deg, int n) {
  int i = blockIdx.x * blockDim.x + threadIdx.x;
  if (i < n) {
    float d = deg[i];
    deg[i] = (d > 0.0f) ? rsqrtf(d) : 0.0f;
  }
}

// int64 edge list -> int32 src/dst + fused per-edge norm (read once per layer
// instead of recomputing 128/237 times per edge in the scatter kernels).
__global__ void edge_prep_kernel(const long long* __restrict__ ei,
                                 const float* __restrict__ dinv,
                                 int* __restrict__ s32, int* __restrict__ d32,
                                 float* __restrict__ norm, int E) {
  int e = blockIdx.x * blockDim.x + threadIdx.x;
  if (e >= E) return;
  long long s = ei[e];
  long long d = ei[(size_t)E + e];
  s32[e] = (int)s;
  d32[e] = (int)d;
  norm[e] = dinv[s] * dinv[d];
}

// C[M x N] = A[M x K] @ B[K x N] via V_WMMA_F32_16X16X4_F32.
// Block = 256 threads (8 waves). The block's B panel (K x NT*16 columns,
// zero-padded past N) is staged in LDS with the 4 K-values of each column
// stored contiguously:  lds[(kk*NC + nl)*4 + j] = B[4kk+j][n0+nl].
// A lane {half,r} fragment for (kk, tile t) is one ds_load_b64 at dword
// offset (kk*NC + t*16 + r)*4 + 2*half -> 16-lane stride of 4 dwords covers
// all 64 banks, half offset interleaves: conflict-free.
// Each wave grid-strides over 16-row strips of A. GUARD=false when every
// column of the panel is provably < N (layer 1), eliminating EXEC masking.
template <int K, int NT, bool GUARD>
__global__ __launch_bounds__(256) void gemm_wmma_lds(
    const float* __restrict__ A, const float* __restrict__ B,
    float* __restrict__ C, int N) {
  constexpr int KK = K / 4;
  constexpr int NC = NT * 16;
  extern __shared__ float ldsB[];  // KK * NC * 4 floats

  const int tid = threadIdx.x;
  const int n0  = blockIdx.y * NC;

  // ---- stage + swizzle B panel ----
  for (int idx = tid; idx < KK * NC; idx += 256) {
    int kk = idx / NC;
    int nl = idx - kk * NC;
    int n  = n0 + nl;
    float4 v = {0.f, 0.f, 0.f, 0.f};
    if (!GUARD || n < N) {
      const float* bp = B + (size_t)(4 * kk) * N + n;
      v.x = bp[0];
      v.y = bp[(size_t)N];
      v.z = bp[2 * (size_t)N];
      v.w = bp[3 * (size_t)N];
    }
    *(float4*)(ldsB + (size_t)idx * 4) = v;
  }
  __syncthreads();

  const int lane = tid & 31;
  const int wave = tid >> 5;
  const int half = lane >> 4;
  const int r    = lane & 15;
  const float* __restrict__ lb = ldsB + 2 * half;

  for (int strip = blockIdx.x * 8 + wave; strip < M_STRIPS;
       strip += gridDim.x * 8) {
    const int m0 = strip * 16;
    v8f acc[NT] = {};
    const float* __restrict__ arow = A + (size_t)(m0 + r) * K + 2 * half;

    // Speculative RT prefetch of next strip's A row (global_prefetch_b8).
    int nstrip = strip + gridDim.x * 8;
    if (nstrip < M_STRIPS)
      __builtin_prefetch(A + (size_t)(nstrip * 16 + r) * K, 0, 1);

#pragma unroll 4
    for (int kk = 0; kk < KK; ++kk) {
      v2f a;
      a.x = arow[4 * kk + 0];
      a.y = arow[4 * kk + 1];
#pragma unroll
      for (int t = 0; t < NT; ++t) {
        v2f b = *(const v2f*)(lb + (size_t)(kk * NC + t * 16 + r) * 4);
        acc[t] = __builtin_amdgcn_wmma_f32_16x16x4_f32(
            /*neg_a=*/false, a, /*neg_b=*/false, b,
            /*c_mod=*/(short)0, acc[t], /*reuse_a=*/false, /*reuse_b=*/false);
      }
    }

    // C/D layout: VGPR v -> row m0 + 8*half + v, col n0 + t*16 + r.
    float* crow0 = C + (size_t)(m0 + 8 * half) * N;
#pragma unroll
    for (int t = 0; t < NT; ++t) {
      int n = n0 + t * 16 + r;
      if (!GUARD || n < N) {
        float* cp = crow0 + n;
#pragma unroll
        for (int v = 0; v < 8; ++v) {
          *cp = acc[t][v];
          cp += N;
        }
      }
    }
  }
}

// acc[dst, f] += hw[src, f] * norm[e]; one (edge, feature) per thread.
// hw / acc are L2-resident (51-95 MB vs 192 MB L2) so these ~584M
// global_atomic_add_f32 hit the L2 atomic units, not HBM.
template <int F>
__global__ void scatter_kernel(const float* __restrict__ hw,
                               const int* __restrict__ s32,
                               const int* __restrict__ d32,
                               const float* __restrict__ norm,
                               float* __restrict__ acc, int E) {
  int e = blockIdx.x * blockDim.y + threadIdx.y;
  int f = threadIdx.x;
  if (e >= E || f >= F) return;
  int s = s32[e];
  int d = d32[e];
  float nrm = norm[e];
  atomicAdd(&acc[(size_t)d * F + f], hw[(size_t)s * F + f] * nrm);
}

// out[i,f] = acc[i,f] + hw[i,f]*dinv[i]^2 + bias[f]  (self-loop folded in),
// optional ReLU, in place on acc.
template <int F>
__global__ void finalize_kernel(float* __restrict__ acc,
                                const float* __restrict__ hw,
                                const float* __restrict__ dinv,
                                const float* __restrict__ bias, int total,
                                int do_relu) {
  int i = blockIdx.x * blockDim.x + threadIdx.x;
  if (i >= total) return;
  int node = i / F;
  int f    = i - node * F;
  float di = dinv[node];
  float v  = acc[i] + hw[i] * di * di + bias[f];
  if (do_relu) v = v > 0.0f ? v : 0.0f;
  acc[i] = v;
}

extern "C" void kernel_launch(void* const* d_in, const int* in_sizes, int n_in,
                              void* d_out, int out_size, void* d_ws,
                              size_t ws_size, hipStream_t stream) {
  const float*     x  = (const float*)d_in[0];
  const long long* ei = (const long long*)d_in[1];  // int64 edge_index [2, E]
  const float*     W1 = (const float*)d_in[2];
  const float*     b1 = (const float*)d_in[3];
  const float*     W2 = (const float*)d_in[4];
  const float*     b2 = (const float*)d_in[5];
  float* out = (float*)d_out;

  // Workspace: dinv | s32 | d32 | norm | hw1 | h | hw2   (~217 MB)
  float* dinv = (float*)d_ws;
  int*   s32  = (int*)(dinv + 100352);
  int*   d32  = s32 + N_EDGES;
  float* norm = (float*)(d32 + N_EDGES);
  float* hw1  = norm + N_EDGES;
  float* h    = hw1 + (size_t)N_NODES * HID_DIM;
  float* hw2  = h + (size_t)N_NODES * HID_DIM;

  // --- gcn_norm: deg = 1 + indegree, dinv = rsqrt(deg), per-edge norm ---
  init_ones_kernel<<<(N_NODES + 255) / 256, 256, 0, stream>>>(dinv, N_NODES);
  degree_kernel<<<(N_EDGES + 255) / 256, 256, 0, stream>>>(ei, dinv, N_EDGES);
  rsqrt_kernel<<<(N_NODES + 255) / 256, 256, 0, stream>>>(dinv, N_NODES);
  edge_prep_kernel<<<(N_EDGES + 255) / 256, 256, 0, stream>>>(
      ei, dinv, s32, d32, norm, N_EDGES);

  const int gx = (M_STRIPS + 7) / 8;  // 782 blocks, 8 waves each

  // --- layer 1: hw1 = x @ W1 ; 2 n-groups x NT=4 -> 64 KB LDS, no guard ---
  gemm_wmma_lds<IN_DIM, 4, false>
      <<<dim3(gx, 2), 256, (IN_DIM / 4) * 64 * 4 * sizeof(float), stream>>>(
          x, W1, hw1, HID_DIM);
  hipMemsetAsync(h, 0, (size_t)N_NODES * HID_DIM * sizeof(float), stream);
  scatter_kernel<HID_DIM><<<dim3((N_EDGES + 1) / 2), dim3(HID_DIM, 2), 0,
                            stream>>>(hw1, s32, d32, norm, h, N_EDGES);
  finalize_kernel<HID_DIM><<<(N_NODES * HID_DIM + 255) / 256, 256, 0, stream>>>(
      h, hw1, dinv, b1, N_NODES * HID_DIM, 1);

  // --- layer 2: hw2 = h @ W2 ; 3 n-groups x NT=5 -> 40 KB LDS, guarded ---
  gemm_wmma_lds<HID_DIM, 5, true>
      <<<dim3(gx, 3), 256, (HID_DIM / 4) * 80 * 4 * sizeof(float), stream>>>(
          h, W2, hw2, OUT_DIM);
  hipMemsetAsync(out, 0, (size_t)N_NODES * OUT_DIM * sizeof(float), stream);
  scatter_kernel<OUT_DIM><<<dim3(N_EDGES), dim3(256, 1), 0, stream>>>(
      hw2, s32, d32, norm, out, N_EDGES);
  finalize_kernel<OUT_DIM><<<(N_NODES * OUT_DIM + 255) / 256, 256, 0, stream>>>(
      out, hw2, dinv, b2, N_NODES * OUT_DIM, 0);
}